// GCN_89567247991122
// MI455X (gfx1250) — compile-verified
//
#include <hip/hip_runtime.h>
#include <stdint.h>

typedef float v2f __attribute__((ext_vector_type(2)));
typedef float v8f __attribute__((ext_vector_type(8)));

#define FIN 256
#define H1  16
#define C2  40

// ---------------------------------------------------------------- degree ----
__global__ void __launch_bounds__(256) gcn_deg_init(float* __restrict__ deg, int n) {
    int i = blockIdx.x * blockDim.x + threadIdx.x;
    if (i < n) deg[i] = 1.0f;                       // self-loop contribution
}

__global__ void __launch_bounds__(256) gcn_deg_scatter(const int* __restrict__ cols,
                                                       float* __restrict__ deg, int e) {
    int i = blockIdx.x * blockDim.x + threadIdx.x;
    if (i < e) unsafeAtomicAdd(&deg[cols[i]], 1.0f);
}

__global__ void __launch_bounds__(256) gcn_deg_finalize(float* __restrict__ deg, int n) {
    int i = blockIdx.x * blockDim.x + threadIdx.x;
    if (i < n) {
        float d = deg[i];
        deg[i] = (d > 0.0f) ? rsqrtf(d) : 0.0f;     // in-place -> dinv
    }
}

// ------------------------------------------------------- GEMM1: hs = dinv*(x@W1)
// One wave32 per 16-row tile. K=256 consumed as 64 x V_WMMA_F32_16X16X4_F32.
__global__ void __launch_bounds__(256) gcn_gemm1(const float* __restrict__ x,
                                                 const float* __restrict__ w1,
                                                 const float* __restrict__ dinv,
                                                 float* __restrict__ hs,
                                                 float* __restrict__ S,
                                                 int n, int ntiles) {
    __shared__ float sw1[FIN * H1];                 // 16 KB of 320 KB WGP LDS
    for (int i = threadIdx.x; i < FIN * H1; i += blockDim.x) sw1[i] = w1[i];
    __syncthreads();

    const int wave = threadIdx.x >> 5;
    const int lane = threadIdx.x & 31;
    const int tile = blockIdx.x * (blockDim.x >> 5) + wave;
    if (tile >= ntiles) return;                     // whole-wave exit: EXEC stays full

    const int laneLo = lane & 15;
    const int hi     = lane >> 4;                   // 0: K+{0,1}  1: K+{2,3}
    const int row    = tile * 16 + laneLo;
    const int rowc   = (row < n) ? row : (n - 1);   // clamp (only matters for tail)

    const float* __restrict__ xrow = x + (size_t)rowc * FIN + (hi ? 2 : 0);

    v8f acc = {};
#pragma unroll 8
    for (int kk = 0; kk < FIN / 4; ++kk) {
        // A 16x4 f32 fragment: V0=K(0|2), V1=K(1|3) -> contiguous float2 per lane
        v2f a = *(const v2f*)(xrow + kk * 4);
        // B 4x16 f32 fragment from LDS: V0 = row(kb), V1 = row(kb+1), lane-half split
        const int kb = kk * 4 + (hi ? 2 : 0);
        v2f b;
        b.x = sw1[(kb + 0) * H1 + laneLo];
        b.y = sw1[(kb + 1) * H1 + laneLo];
        acc = __builtin_amdgcn_wmma_f32_16x16x4_f32(false, a, false, b,
                                                    (short)0, acc, false, false);
    }

    // C tile: VGPR j <-> rows (j, j+8) split across lane halves.
    // hs = dinv[row] * h1;  S seeded with hs (self-loop term of the scatter sum).
    const int mbase = tile * 16 + hi * 8;
    if (tile * 16 + 16 <= n) {
        // full tile: unguarded, constant-offset stores
        float* __restrict__ ph = hs + (size_t)mbase * H1 + laneLo;
        float* __restrict__ pS = S  + (size_t)mbase * H1 + laneLo;
        const float* __restrict__ pd = dinv + mbase;
#pragma unroll
        for (int j = 0; j < 8; ++j) {
            float val = acc[j] * pd[j];
            ph[j * H1] = val;
            pS[j * H1] = val;
        }
    } else {
#pragma unroll
        for (int j = 0; j < 8; ++j) {
            int m = mbase + j;
            if (m < n) {
                float val = acc[j] * dinv[m];
                hs[(size_t)m * H1 + laneLo] = val;
                S [(size_t)m * H1 + laneLo] = val;
            }
        }
    }
}

// ------------------------------------------------- scatter1: S[col] += hs[row]
__global__ void __launch_bounds__(256) gcn_scatter16(const int* __restrict__ rows,
                                                     const int* __restrict__ cols,
                                                     const float* __restrict__ hs,
                                                     float* __restrict__ S,
                                                     unsigned total) {
    unsigned t = blockIdx.x * blockDim.x + threadIdx.x;
    if (t >= total) return;
    unsigned e = t >> 4;
    unsigned f = t & 15u;
    int r = rows[e], c = cols[e];
    unsafeAtomicAdd(&S[(size_t)c * H1 + f], hs[(size_t)r * H1 + f]);
}

// ------------- GEMM2 (fused): a1 = relu(dinv*S + b1); hs2 = dinv*(a1 @ W2) ---
__global__ void __launch_bounds__(256) gcn_gemm2(const float* __restrict__ S,
                                                 const float* __restrict__ w2,
                                                 const float* __restrict__ b1,
                                                 const float* __restrict__ dinv,
                                                 float* __restrict__ hs2,
                                                 float* __restrict__ S2,
                                                 int n, int ntiles) {
    __shared__ float sw2[H1 * C2];                  // 640 floats
    __shared__ float sb1[H1];
    for (int i = threadIdx.x; i < H1 * C2; i += blockDim.x) sw2[i] = w2[i];
    if (threadIdx.x < H1) sb1[threadIdx.x] = b1[threadIdx.x];
    __syncthreads();

    const int wave = threadIdx.x >> 5;
    const int lane = threadIdx.x & 31;
    const int tile = blockIdx.x * (blockDim.x >> 5) + wave;
    if (tile >= ntiles) return;

    const int laneLo = lane & 15;
    const int hi     = lane >> 4;
    const int row    = tile * 16 + laneLo;
    const int rowc   = (row < n) ? row : (n - 1);
    const float dv   = dinv[rowc];
    const bool  full = (tile * 16 + 16 <= n);

    // A fragments (K=16 -> 4 chunks), with bias+ReLU+prescale fused in
    v2f afrag[4];
#pragma unroll
    for (int kc = 0; kc < 4; ++kc) {
        int c0 = kc * 4 + hi * 2;
        float s0 = S[(size_t)rowc * H1 + c0];
        float s1 = S[(size_t)rowc * H1 + c0 + 1];
        afrag[kc].x = fmaxf(fmaf(dv, s0, sb1[c0]),     0.0f);
        afrag[kc].y = fmaxf(fmaf(dv, s1, sb1[c0 + 1]), 0.0f);
    }

    const int mbase = tile * 16 + hi * 8;
#pragma unroll
    for (int nt = 0; nt < 3; ++nt) {                // N = 40 -> tiles 16,16,8
        int ncol    = nt * 16 + laneLo;
        float nmask = (ncol < C2) ? 1.0f : 0.0f;    // value-masked, EXEC stays full
        int nidx    = (ncol < C2) ? ncol : (C2 - 1);
        v8f acc = {};
#pragma unroll
        for (int kc = 0; kc < 4; ++kc) {
            int kb = kc * 4 + hi * 2;
            v2f b;
            b.x = sw2[(kb + 0) * C2 + nidx] * nmask;
            b.y = sw2[(kb + 1) * C2 + nidx] * nmask;
            acc = __builtin_amdgcn_wmma_f32_16x16x4_f32(false, afrag[kc], false, b,
                                                        (short)0, acc, false, false);
        }
        if (full && nt < 2) {
            // full tile, full 16-wide N tile: unguarded constant-offset stores
            float* __restrict__ ph = hs2 + (size_t)mbase * C2 + ncol;
            float* __restrict__ pS = S2  + (size_t)mbase * C2 + ncol;
            const float* __restrict__ pd = dinv + mbase;
#pragma unroll
            for (int j = 0; j < 8; ++j) {
                float val = acc[j] * pd[j];
                ph[j * C2] = val;
                pS[j * C2] = val;       // self-loop seed
            }
        } else if (ncol < C2) {
#pragma unroll
            for (int j = 0; j < 8; ++j) {
                int m = mbase + j;
                if (m < n) {
                    float val = acc[j] * dinv[m];
                    hs2[(size_t)m * C2 + ncol] = val;
                    S2 [(size_t)m * C2 + ncol] = val;
                }
            }
        }
    }
}

// ----------------------------------------------- scatter2: S2[col] += hs2[row]
__global__ void __launch_bounds__(256) gcn_scatter40(const int* __restrict__ rows,
                                                     const int* __restrict__ cols,
                                                     const float* __restrict__ hs2,
                                                     float* __restrict__ S2,
                                                     unsigned total) {
    unsigned t = blockIdx.x * blockDim.x + threadIdx.x;
    if (t >= total) return;
    unsigned e = t / 40u;
    unsigned f = t - e * 40u;
    int r = rows[e], c = cols[e];
    unsafeAtomicAdd(&S2[(size_t)c * C2 + f], hs2[(size_t)r * C2 + f]);
}

// ----------------------- out = log_softmax(dinv*S2 + b2): one wave32 per node
__global__ void __launch_bounds__(256) gcn_logsoftmax(const float* __restrict__ S2,
                                                      const float* __restrict__ dinv,
                                                      const float* __restrict__ b2,
                                                      float* __restrict__ out,
                                                      int n) {
    const int lane = threadIdx.x & 31;
    const int node = (blockIdx.x * blockDim.x + threadIdx.x) >> 5;
    if (node >= n) return;

    const float dv = dinv[node];
    const float* __restrict__ z = S2 + (size_t)node * C2;

    float z1 = fmaf(dv, z[lane], b2[lane]);                         // classes 0..31
    float z2 = (lane < 8) ? fmaf(dv, z[32 + lane], b2[32 + lane])   // classes 32..39
                          : -3.0e38f;

    float m = fmaxf(z1, z2);
#pragma unroll
    for (int off = 16; off > 0; off >>= 1)
        m = fmaxf(m, __shfl_xor(m, off, 32));

    float s = __expf(z1 - m) + ((lane < 8) ? __expf(z2 - m) : 0.0f);
#pragma unroll
    for (int off = 16; off > 0; off >>= 1)
        s += __shfl_xor(s, off, 32);

    float lse = m + __logf(s);
    out[(size_t)node * C2 + lane] = z1 - lse;
    if (lane < 8) out[(size_t)node * C2 + 32 + lane] = z2 - lse;
}

// ---------------------------------------------------------------------------
extern "C" void kernel_launch(void* const* d_in, const int* in_sizes, int n_in,
                              void* d_out, int out_size, void* d_ws, size_t ws_size,
                              hipStream_t stream) {
    const float* x   = (const float*)d_in[0];
    const float* W1  = (const float*)d_in[1];
    const float* b1  = (const float*)d_in[2];
    const float* W2  = (const float*)d_in[3];
    const float* b2  = (const float*)d_in[4];
    const int*   ei  = (const int*)d_in[5];

    const int N = in_sizes[0] / FIN;
    const int E = in_sizes[5] / 2;
    const int* rows = ei;         // edge_index[0]
    const int* cols = ei + E;     // edge_index[1]

    // workspace layout (floats): dinv | hs | S | hs2 | S2  = 113*N floats (~90 MB)
    float* ws   = (float*)d_ws;
    float* dinv = ws;
    float* hs   = dinv + (size_t)N;
    float* S    = hs   + (size_t)N * H1;
    float* hs2  = S    + (size_t)N * H1;
    float* S2   = hs2  + (size_t)N * C2;

    float* out = (float*)d_out;

    const int ntiles = (N + 15) / 16;
    const int gemmBlocks = (ntiles + 7) / 8;        // 8 waves per 256-thread block

    gcn_deg_init    <<<(N + 255) / 256, 256, 0, stream>>>(dinv, N);
    gcn_deg_scatter <<<(E + 255) / 256, 256, 0, stream>>>(cols, dinv, E);
    gcn_deg_finalize<<<(N + 255) / 256, 256, 0, stream>>>(dinv, N);

    gcn_gemm1<<<gemmBlocks, 256, 0, stream>>>(x, W1, dinv, hs, S, N, ntiles);

    {
        unsigned total = (unsigned)E * 16u;
        gcn_scatter16<<<(total + 255u) / 256u, 256, 0, stream>>>(rows, cols, hs, S, total);
    }

    gcn_gemm2<<<gemmBlocks, 256, 0, stream>>>(S, W2, b1, dinv, hs2, S2, N, ntiles);

    {
        unsigned total = (unsigned)E * 40u;
        gcn_scatter40<<<(total + 255u) / 256u, 256, 0, stream>>>(rows, cols, hs2, S2, total);
    }

    gcn_logsoftmax<<<((unsigned)N * 32u + 255u) / 256u, 256, 0, stream>>>(S2, dinv, b2, out, N);
}